// ResBlock_71545565217396
// MI455X (gfx1250) — compile-verified
//
#include <hip/hip_runtime.h>
#include <hip/hip_bf16.h>

typedef float v2f __attribute__((ext_vector_type(2)));
typedef float v8f __attribute__((ext_vector_type(8)));

#define NPIX_PER_IMG 4096   // H*W = 64*64
#define NCH 64
#define NIMG 16
#define BN_EPS 1e-5f

// ---------------------------------------------------------------------------
// Kernel 1: y1[b,o,hw] = -sum_c |x[b,c,hw] - w1[o,c]|
// thread = pixel (b,hw). x channel vector kept in 64 VGPRs (wave32, plenty of
// registers). w[o*64+c] is wave-uniform -> compiler emits s_load (SMEM), so
// the inner body is pure VALU: v_sub_f32 + v_add_f32 with |src| modifier.
// ---------------------------------------------------------------------------
__global__ __launch_bounds__(256) void adder_kernel(
    const float* __restrict__ x, const float* __restrict__ w,
    float* __restrict__ y)
{
    const int t  = blockIdx.x * 256 + threadIdx.x;   // 0..65535
    const int b  = t >> 12;                          // image
    const int hw = t & 4095;                         // pixel in image
    const float* xp = x + (size_t)b * NCH * NPIX_PER_IMG + hw;

    float xr[NCH];
#pragma unroll
    for (int c = 0; c < NCH; ++c) xr[c] = xp[(size_t)c * NPIX_PER_IMG];

    float* yp = y + (size_t)b * NCH * NPIX_PER_IMG + hw;
#pragma unroll 2
    for (int o = 0; o < NCH; ++o) {
        const float* wo = w + o * NCH;               // uniform -> scalar loads
        float a = 0.f;
#pragma unroll
        for (int c = 0; c < NCH; ++c) a += __builtin_fabsf(xr[c] - wo[c]);
        yp[(size_t)o * NPIX_PER_IMG] = -a;
    }
}

// ---------------------------------------------------------------------------
// Kernel 3: h = relu(bn1(y1)) on the fly, then adder with w2.
// bn1 is pre-folded into a[c], b[c] (ab1 buffer): h = max(0, v*a[c] + b[c]).
// ---------------------------------------------------------------------------
__global__ __launch_bounds__(256) void adder_bnrelu_kernel(
    const float* __restrict__ y1, const float* __restrict__ ab1,
    const float* __restrict__ w, float* __restrict__ y2)
{
    const int t  = blockIdx.x * 256 + threadIdx.x;
    const int b  = t >> 12;
    const int hw = t & 4095;
    const float* p = y1 + (size_t)b * NCH * NPIX_PER_IMG + hw;

    float xr[NCH];
#pragma unroll
    for (int c = 0; c < NCH; ++c) {
        float v = p[(size_t)c * NPIX_PER_IMG];
        xr[c] = fmaxf(0.f, v * ab1[c] + ab1[NCH + c]);   // ab1: uniform s_loads
    }

    float* yp = y2 + (size_t)b * NCH * NPIX_PER_IMG + hw;
#pragma unroll 2
    for (int o = 0; o < NCH; ++o) {
        const float* wo = w + o * NCH;
        float a = 0.f;
#pragma unroll
        for (int c = 0; c < NCH; ++c) a += __builtin_fabsf(xr[c] - wo[c]);
        yp[(size_t)o * NPIX_PER_IMG] = -a;
    }
}

// ---------------------------------------------------------------------------
// Kernel 2/4: per-channel mean/var via V_WMMA_F32_16X16X4_F32 reduction.
// Block = one channel. 16 waves <-> 16 images; each wave covers its 4096-float
// channel slice, feeding 64 elements (float2/lane) per pair of WMMAs with an
// all-ones A matrix. Summing all 8 C VGPRs over all 32 lanes yields exactly
// 16x the slice sum regardless of the hardware (lane,VGPR)->(k,n) mapping.
// Finalizes BN constants a[c]=gamma*rsqrt(var+eps), b[c]=beta-mean*a.
// ---------------------------------------------------------------------------
__global__ __launch_bounds__(512) void stats_bn_kernel(
    const float* __restrict__ y, const float* __restrict__ gamma,
    const float* __restrict__ beta, float* __restrict__ ab)
{
    const int ch   = blockIdx.x;        // 0..63
    const int tid  = threadIdx.x;       // 0..511
    const int wid  = tid >> 5;          // wave id == image id, 0..15
    const int lane = tid & 31;

    const size_t base = ((size_t)wid * NCH + ch) * NPIX_PER_IMG;

    v2f ones = {1.0f, 1.0f};
    v8f cs = {0.f, 0.f, 0.f, 0.f, 0.f, 0.f, 0.f, 0.f};
    v8f cq = {0.f, 0.f, 0.f, 0.f, 0.f, 0.f, 0.f, 0.f};

#pragma unroll 4
    for (int it = 0; it < 64; ++it) {
        const float2 v = *(const float2*)(y + base + it * 64 + lane * 2);
        v2f bm;  bm[0] = v.x;        bm[1] = v.y;
        v2f bm2; bm2[0] = v.x * v.x; bm2[1] = v.y * v.y;
        // D = A(ones,16x4) * B(4x16) + C  -> column sums accumulate in C
        cs = __builtin_amdgcn_wmma_f32_16x16x4_f32(
                false, ones, false, bm,  (short)0, cs, false, false);
        cq = __builtin_amdgcn_wmma_f32_16x16x4_f32(
                false, ones, false, bm2, (short)0, cq, false, false);
    }

    float s = cs[0] + cs[1] + cs[2] + cs[3] + cs[4] + cs[5] + cs[6] + cs[7];
    float q = cq[0] + cq[1] + cq[2] + cq[3] + cq[4] + cq[5] + cq[6] + cq[7];
#pragma unroll
    for (int off = 16; off > 0; off >>= 1) {
        s += __shfl_xor(s, off);
        q += __shfl_xor(q, off);
    }

    __shared__ float ls[16], lq[16];
    if (lane == 0) { ls[wid] = s; lq[wid] = q; }
    __syncthreads();

    if (tid == 0) {
        float S = 0.f, Q = 0.f;
#pragma unroll
        for (int i = 0; i < 16; ++i) { S += ls[i]; Q += lq[i]; }
        // every element was counted 16x (16 replicated D rows)
        const float invN = 1.0f / (16.0f * (float)(NIMG * NPIX_PER_IMG));
        const float mean = S * invN;
        const float var  = Q * invN - mean * mean;
        const float a    = gamma[ch] * rsqrtf(var + BN_EPS);
        ab[ch]       = a;
        ab[NCH + ch] = beta[ch] - mean * a;
    }
}

// ---------------------------------------------------------------------------
// Kernel 5: out = relu(bn2(y2) + x), vectorized float4. A 256-thread block
// covers 1024 elements, a quarter of one channel plane -> ab2 reads are uniform.
// ---------------------------------------------------------------------------
__global__ __launch_bounds__(256) void resid_kernel(
    const float* __restrict__ y2, const float* __restrict__ x,
    const float* __restrict__ ab2, float* __restrict__ out)
{
    const size_t i = ((size_t)blockIdx.x * 256 + threadIdx.x) * 4;
    const int ch = (int)((i >> 12) & 63);
    const float a = ab2[ch], b = ab2[NCH + ch];
    const float4 v  = *(const float4*)(y2 + i);
    const float4 xv = *(const float4*)(x + i);
    float4 r;
    r.x = fmaxf(0.f, v.x * a + b + xv.x);
    r.y = fmaxf(0.f, v.y * a + b + xv.y);
    r.z = fmaxf(0.f, v.z * a + b + xv.z);
    r.w = fmaxf(0.f, v.w * a + b + xv.w);
    *(float4*)(out + i) = r;
}

// ---------------------------------------------------------------------------
extern "C" void kernel_launch(void* const* d_in, const int* in_sizes, int n_in,
                              void* d_out, int out_size, void* d_ws, size_t ws_size,
                              hipStream_t stream) {
    const float* x   = (const float*)d_in[0];
    const float* w1  = (const float*)d_in[1];
    const float* g1  = (const float*)d_in[2];
    const float* be1 = (const float*)d_in[3];
    const float* w2  = (const float*)d_in[4];
    const float* g2  = (const float*)d_in[5];
    const float* be2 = (const float*)d_in[6];
    float* out = (float*)d_out;

    char* ws  = (char*)d_ws;
    float* y1  = (float*)ws;                               // 16 MB
    float* y2  = (float*)(ws + (size_t)16 * 1024 * 1024);  // 16 MB
    float* ab1 = (float*)(ws + (size_t)32 * 1024 * 1024);  // 128 floats
    float* ab2 = ab1 + 128;                                // 128 floats

    adder_kernel       <<<256, 256, 0, stream>>>(x, w1, y1);
    stats_bn_kernel    <<<64, 512, 0, stream>>>(y1, g1, be1, ab1);
    adder_bnrelu_kernel<<<256, 256, 0, stream>>>(y1, ab1, w2, y2);
    stats_bn_kernel    <<<64, 512, 0, stream>>>(y2, g2, be2, ab2);
    resid_kernel       <<<4096, 256, 0, stream>>>(y2, x, ab2, out);
}